// Teacher_4269197492518
// MI455X (gfx1250) — compile-verified
//
#include <hip/hip_runtime.h>
#include <hip/hip_bf16.h>

// ---------------------------------------------------------------------------
// VGAE encoder: z = mu + eps * exp(min(logstd, 10))
//   mu     = GCN(x, W_mu, b_mu)
//   logstd = GCN(x, W_logstd, b_logstd)
// GCN(x,W,b)[d] = dinv[d] * ( g[d] + sum_{e: dst(e)=d} g[src(e)] ) + b
//   where g[i] = dinv[i] * (x[i] @ W),  dinv[i] = rsqrt(1 + indegree[i])
//
// Phases (roofline): dual fp32-WMMA GEMM (~100 MB traffic, compute-trivial),
// edge scatter (dominant: 205M L2 float atomics + 1.6GB gathers, L2-resident),
// fused elementwise finalize (~100 MB streamed, float4-vectorized).
// ---------------------------------------------------------------------------

#define IN_DIM  256
#define OUT_DIM 128

typedef __attribute__((ext_vector_type(2))) float v2f;
typedef __attribute__((ext_vector_type(8))) float v8f;

// --------------------------- degree kernels --------------------------------

__global__ void deg_init_kernel(float* __restrict__ deg, int n) {
    int i = blockIdx.x * blockDim.x + threadIdx.x;
    if (i < n) deg[i] = 1.0f;  // self-loop
}

__global__ void deg_count_kernel(const int* __restrict__ ei, float* __restrict__ deg,
                                 int E) {
    int e = blockIdx.x * blockDim.x + threadIdx.x;
    if (e < E) {
        int d = ei[E + e];  // dst row of edge_index [2, E]
        atomicAdd(&deg[d], 1.0f);
    }
}

// --------------------------- fused dual GEMM (fp32 WMMA) -------------------
// One wave per 16x16 output tile. K=256 in steps of 4 via V_WMMA_F32_16X16X4_F32.
// Computes h_mu and h_logstd sharing the A-matrix (x) loads, scales by
// dinv[row] and writes g (gather source) and s (accumulator, init = g).
//
// A 16x4 fp32 layout (ISA 7.12.2): lanes 0-15 hold M=lane, lanes 16-31 hold
// M=lane-16; VGPR0 = {K, K+2}, VGPR1 = {K+1, K+3} by lane half.
// B 4x16 mirrors A (rows striped across lane-halves, col = lane%16).
// C/D: VGPR v holds row M = v + 8*(lane>=16), col N = lane%16.

__global__ void __launch_bounds__(256)
gcn_dual_gemm_wmma(const float* __restrict__ x,
                   const float* __restrict__ Wmu,
                   const float* __restrict__ Wls,
                   const float* __restrict__ deg,
                   float* __restrict__ g_mu, float* __restrict__ s_mu,
                   float* __restrict__ g_ls, float* __restrict__ s_ls,
                   int n) {
    const int lane = threadIdx.x & 31;
    const int wave = threadIdx.x >> 5;          // 8 waves/block
    const int tile = blockIdx.x * 8 + wave;     // mtiles*8 tiles total
    const int mt   = tile >> 3;                 // row tile
    const int nt   = tile & 7;                  // col tile (OUT_DIM/16 == 8)
    const int row0 = mt * 16;
    const int col0 = nt * 16;

    const int half = lane >> 4;                 // 0: lanes 0-15, 1: lanes 16-31
    const int l16  = lane & 15;

    // Clamp the A-row read index so partial tiles keep EXEC all-ones through
    // the WMMA loop (stores are guarded separately).
    int arow = row0 + l16;
    if (arow > n - 1) arow = n - 1;
    const float* __restrict__ xrow = x + (long long)arow * IN_DIM;

    v8f acc_mu = {};
    v8f acc_ls = {};

    for (int k = 0; k < IN_DIM; k += 4) {
        const int kk = k + half * 2;
        v2f a;
        a.x = xrow[kk];
        a.y = xrow[kk + 1];
        v2f bm, bl;
        bm.x = Wmu[(kk)     * OUT_DIM + col0 + l16];
        bm.y = Wmu[(kk + 1) * OUT_DIM + col0 + l16];
        bl.x = Wls[(kk)     * OUT_DIM + col0 + l16];
        bl.y = Wls[(kk + 1) * OUT_DIM + col0 + l16];
        acc_mu = __builtin_amdgcn_wmma_f32_16x16x4_f32(
            false, a, false, bm, (short)0, acc_mu, false, false);
        acc_ls = __builtin_amdgcn_wmma_f32_16x16x4_f32(
            false, a, false, bl, (short)0, acc_ls, false, false);
    }

    // Write back g = dinv * h, and initialize accumulator s = g.
    if (row0 + 16 <= n) {
        // Uniform full-tile fast path: straight-line, no per-row exec masking.
#pragma unroll
        for (int v = 0; v < 8; ++v) {
            const int r = row0 + v + half * 8;
            const float dinv = rsqrtf(deg[r]);
            const float gm = acc_mu[v] * dinv;
            const float gl = acc_ls[v] * dinv;
            const long long o = (long long)r * OUT_DIM + col0 + l16;
            g_mu[o] = gm;  s_mu[o] = gm;
            g_ls[o] = gl;  s_ls[o] = gl;
        }
    } else {
#pragma unroll
        for (int v = 0; v < 8; ++v) {
            const int r = row0 + v + half * 8;
            if (r < n) {
                const float dinv = rsqrtf(deg[r]);
                const float gm = acc_mu[v] * dinv;
                const float gl = acc_ls[v] * dinv;
                const long long o = (long long)r * OUT_DIM + col0 + l16;
                g_mu[o] = gm;  s_mu[o] = gm;
                g_ls[o] = gl;  s_ls[o] = gl;
            }
        }
    }
}

// --------------------------- edge scatter ----------------------------------
// One wave (32 lanes) per edge; each lane handles 4 consecutive features via
// a float4 gather (L2-resident) + 4 float atomics, for both mu and logstd.
// Edge indices are wave-uniform -> broadcast loads; each wave's atomic
// footprint is 2 contiguous 512B rows.

__global__ void edge_scatter_kernel(const int* __restrict__ ei,
                                    const float* __restrict__ g_mu,
                                    const float* __restrict__ g_ls,
                                    float* __restrict__ s_mu,
                                    float* __restrict__ s_ls,
                                    int E) {
    const long long gid = (long long)blockIdx.x * blockDim.x + threadIdx.x;
    const int e = (int)(gid >> 5);
    if (e >= E) return;
    const int f = ((int)gid & 31) * 4;  // 32 lanes * 4 floats = 128 features

    const int src = ei[e];
    const int dst = ei[E + e];
    const long long so   = (long long)src * OUT_DIM + f;
    const long long doff = (long long)dst * OUT_DIM + f;

    const float4 gm = *(const float4*)(g_mu + so);
    const float4 gl = *(const float4*)(g_ls + so);

    atomicAdd(&s_mu[doff + 0], gm.x);
    atomicAdd(&s_mu[doff + 1], gm.y);
    atomicAdd(&s_mu[doff + 2], gm.z);
    atomicAdd(&s_mu[doff + 3], gm.w);
    atomicAdd(&s_ls[doff + 0], gl.x);
    atomicAdd(&s_ls[doff + 1], gl.y);
    atomicAdd(&s_ls[doff + 2], gl.z);
    atomicAdd(&s_ls[doff + 3], gl.w);
}

// --------------------------- finalize --------------------------------------
// mu = dinv*s_mu + b_mu ; ls = min(dinv*s_ls + b_ls, 10) ; z = mu + eps*exp(ls)
// float4 per thread: b128 loads/stores, 1 deg load + rsqrt per 4 elements.

__global__ void finalize_kernel(const float* __restrict__ s_mu,
                                const float* __restrict__ s_ls,
                                const float* __restrict__ deg,
                                const float* __restrict__ b_mu,
                                const float* __restrict__ b_ls,
                                const float* __restrict__ eps,
                                float* __restrict__ z,
                                long long total4) {
    const long long q = (long long)blockIdx.x * blockDim.x + threadIdx.x;
    if (q >= total4) return;
    const long long i = q * 4;
    const int r = (int)(i >> 7);        // OUT_DIM == 128
    const int f = (int)(i & 127);
    const float dinv = rsqrtf(deg[r]);

    const float4 sm = *(const float4*)(s_mu + i);
    const float4 sl = *(const float4*)(s_ls + i);
    const float4 bm = *(const float4*)(b_mu + f);
    const float4 bl = *(const float4*)(b_ls + f);
    const float4 ep = *(const float4*)(eps + i);

    float4 out;
    {
        float mu = dinv * sm.x + bm.x;
        float ls = fminf(dinv * sl.x + bl.x, 10.0f);
        out.x = mu + ep.x * expf(ls);
    }
    {
        float mu = dinv * sm.y + bm.y;
        float ls = fminf(dinv * sl.y + bl.y, 10.0f);
        out.y = mu + ep.y * expf(ls);
    }
    {
        float mu = dinv * sm.z + bm.z;
        float ls = fminf(dinv * sl.z + bl.z, 10.0f);
        out.z = mu + ep.z * expf(ls);
    }
    {
        float mu = dinv * sm.w + bm.w;
        float ls = fminf(dinv * sl.w + bl.w, 10.0f);
        out.w = mu + ep.w * expf(ls);
    }
    *(float4*)(z + i) = out;
}

// --------------------------- launch ----------------------------------------

extern "C" void kernel_launch(void* const* d_in, const int* in_sizes, int n_in,
                              void* d_out, int out_size, void* d_ws, size_t ws_size,
                              hipStream_t stream) {
    const float* x     = (const float*)d_in[0];
    const int*   ei    = (const int*)d_in[1];
    const float* W_mu  = (const float*)d_in[2];
    const float* b_mu  = (const float*)d_in[3];
    const float* W_ls  = (const float*)d_in[4];
    const float* b_ls  = (const float*)d_in[5];
    const float* eps   = (const float*)d_in[6];
    float*       z     = (float*)d_out;

    const int n = in_sizes[0] / IN_DIM;   // 50000
    const int E = in_sizes[1] / 2;        // 800000

    // Workspace layout (floats), offsets padded to 64-float (256B) alignment.
    float* ws = (float*)d_ws;
    const long long n_pad  = ((long long)n + 63) & ~63LL;
    const long long nf     = (long long)n * OUT_DIM;
    const long long nf_pad = (nf + 63) & ~63LL;
    float* deg  = ws;
    float* g_mu = deg  + n_pad;
    float* s_mu = g_mu + nf_pad;
    float* g_ls = s_mu + nf_pad;
    float* s_ls = g_ls + nf_pad;
    (void)ws_size; (void)n_in; (void)out_size;

    // 1. degrees (self-loop init + dst counting)
    deg_init_kernel<<<(n + 255) / 256, 256, 0, stream>>>(deg, n);
    deg_count_kernel<<<(E + 255) / 256, 256, 0, stream>>>(ei, deg, E);

    // 2. fused dual GEMM with fp32 WMMA; writes g and s (= g) for both convs
    const int mtiles = (n + 15) / 16;     // 3125 for n=50000
    gcn_dual_gemm_wmma<<<mtiles, 256, 0, stream>>>(
        x, W_mu, W_ls, deg, g_mu, s_mu, g_ls, s_ls, n);

    // 3. edge message scatter (dominant phase): one wave per edge
    const long long sthreads = (long long)E * 32;
    edge_scatter_kernel<<<(int)((sthreads + 255) / 256), 256, 0, stream>>>(
        ei, g_mu, g_ls, s_mu, s_ls, E);

    // 4. finalize: normalize, bias, clamp, reparametrize (float4 vectorized)
    const long long nf4 = nf / 4;
    finalize_kernel<<<(int)((nf4 + 255) / 256), 256, 0, stream>>>(
        s_mu, s_ls, deg, b_mu, b_ls, eps, z, nf4);
}